// SimpleGCN_4638564679932
// MI455X (gfx1250) — compile-verified
//
#include <hip/hip_runtime.h>

#define N_NODES  100000
#define N_EDGES  1600000
#define CH       128
#define N_LAYERS 3
#define N_GRAPHS 512
#define LDS_STRIDE 132   // padded k-stride for transposed W: bank-conflict-free, 8B-aligned

typedef float v2f __attribute__((ext_vector_type(2)));
typedef float v8f __attribute__((ext_vector_type(8)));

__device__ __forceinline__ void atomAddF(float* p, float v) {
    (void)__hip_atomic_fetch_add(p, v, __ATOMIC_RELAXED, __HIP_MEMORY_SCOPE_AGENT);
}

// ---------------------------------------------------------------- degree/norm
__global__ void k_init_deg(float* __restrict__ deg) {
    int i = blockIdx.x * blockDim.x + threadIdx.x;
    if (i < N_NODES) deg[i] = 1.0f;              // self-loop contribution
}

__global__ void k_deg_edges(const int* __restrict__ dst, float* __restrict__ deg) {
    int e = blockIdx.x * blockDim.x + threadIdx.x;
    if (e < N_EDGES) atomAddF(&deg[dst[e]], 1.0f);
}

__global__ void k_make_dinv(float* __restrict__ d) {
    int i = blockIdx.x * blockDim.x + threadIdx.x;
    if (i < N_NODES) {
        float v = d[i];
        d[i] = (v > 0.0f) ? rsqrtf(v) : 0.0f;
    }
}

// ---------------------------------------------------------------- fp32 WMMA GEMM
// T[N x 128] = A[N x 128] @ W[128 x 128], fp32, V_WMMA_F32_16X16X4_F32.
// Block: 256 threads = 8 waves; each wave owns a 16-row strip.
// W is staged TRANSPOSED in LDS (lWt[n][k], stride 132) so each B fragment
// (W[k][n], W[k+1][n]) is one contiguous 8B ds_load_b64, no register shuffles.
__global__ __launch_bounds__(256) void k_gemm_wmma(const float* __restrict__ A,
                                                   const float* __restrict__ W,
                                                   float* __restrict__ T, int n) {
    __shared__ float lWt[CH * LDS_STRIDE];        // 67.6 KB
    const int tid = threadIdx.x;

    // Stage W transposed: coalesced float4 reads of W[k][n0..n0+3], scalar LDS writes.
#pragma unroll
    for (int i = 0; i < 16; ++i) {
        int flat4 = tid + i * 256;                // 4096 float4s
        int k  = flat4 >> 5;                      // row of W
        int n0 = (flat4 & 31) << 2;               // starting col of W
        float4 w = ((const float4*)W)[flat4];
        lWt[(n0 + 0) * LDS_STRIDE + k] = w.x;
        lWt[(n0 + 1) * LDS_STRIDE + k] = w.y;
        lWt[(n0 + 2) * LDS_STRIDE + k] = w.z;
        lWt[(n0 + 3) * LDS_STRIDE + k] = w.w;
    }
    __syncthreads();

    const int wave    = tid >> 5;
    const int lane    = tid & 31;
    const int rowBase = blockIdx.x * 128 + wave * 16;
    if (rowBase >= n) return;                     // wave-uniform: EXEC stays all-1s

    const int m     = lane & 15;                  // row within tile / col within tile
    const int khalf = (lane >> 4) << 1;           // 0 (lanes 0-15) or 2 (lanes 16-31)
    const float* Arow = A + (size_t)(rowBase + m) * CH + khalf;

#pragma unroll
    for (int nt = 0; nt < 8; ++nt) {
        v8f acc = {};
        const int ncol = nt * 16 + m;
        const float* Brow = lWt + ncol * LDS_STRIDE + khalf;
#pragma unroll
        for (int ks = 0; ks < 32; ++ks) {         // K = 128 in steps of 4
            // A 16x4: VGPR0=K0/K2, VGPR1=K1/K3 -> one global b64 per lane
            v2f a = *(const v2f*)(Arow + ks * 4);
            // B 4x16: (K,K+1) contiguous in transposed LDS -> one ds b64 per lane
            v2f b = *(const v2f*)(Brow + ks * 4);
            acc = __builtin_amdgcn_wmma_f32_16x16x4_f32(
                false, a, false, b, (short)0, acc, false, false);
        }
        // D 16x16 fp32: VGPR j -> M = j + 8*(lane>=16), N = lane&15
        float* Tp = T + (size_t)rowBase * CH + ncol;
        const int mhi = (lane >> 4) * 8;
#pragma unroll
        for (int j = 0; j < 8; ++j) Tp[(size_t)(j + mhi) * CH] = acc[j];
    }
}

// ---------------------------------------------------------------- propagation
// h[n][c] = dinv[n]^2 * t[n][c]   (self-loop term; also zero-inits accumulator)
__global__ void k_self_init(const float* __restrict__ t, const float* __restrict__ dinv,
                            float* __restrict__ h) {
    int i = blockIdx.x * blockDim.x + threadIdx.x;   // float4 index over N*32
    if (i >= N_NODES * (CH / 4)) return;
    int node = i >> 5;
    float w = dinv[node] * dinv[node];
    float4 v = ((const float4*)t)[i];
    v.x *= w; v.y *= w; v.z *= w; v.w *= w;
    ((float4*)h)[i] = v;
}

// One wave per edge; lane covers 4 channels; fp32 hardware atomics into h[dst].
__global__ __launch_bounds__(256) void k_scatter(const int* __restrict__ src,
                                                 const int* __restrict__ dst,
                                                 const float* __restrict__ dinv,
                                                 const float* __restrict__ t,
                                                 float* __restrict__ h) {
    int e = blockIdx.x * 8 + (threadIdx.x >> 5);
    if (e >= N_EDGES) return;
    int lane = threadIdx.x & 31;
    int s = src[e], d = dst[e];
    float w = dinv[s] * dinv[d];
    float4 v = ((const float4*)(t + (size_t)s * CH))[lane];
    float* hp = h + (size_t)d * CH + lane * 4;
    atomAddF(hp + 0, w * v.x);
    atomAddF(hp + 1, w * v.y);
    atomAddF(hp + 2, w * v.z);
    atomAddF(hp + 3, w * v.w);
}

__global__ void k_bias_relu(float* __restrict__ h, const float* __restrict__ b) {
    int i = blockIdx.x * blockDim.x + threadIdx.x;   // float4 index over N*32
    if (i >= N_NODES * (CH / 4)) return;
    int c = (i & 31) * 4;
    float4 v = ((float4*)h)[i];
    v.x = fmaxf(v.x + b[c + 0], 0.0f);
    v.y = fmaxf(v.y + b[c + 1], 0.0f);
    v.z = fmaxf(v.z + b[c + 2], 0.0f);
    v.w = fmaxf(v.w + b[c + 3], 0.0f);
    ((float4*)h)[i] = v;
}

// ---------------------------------------------------------------- pool + head
__global__ void k_out_init(float* __restrict__ out, const float* __restrict__ head_b) {
    int g = blockIdx.x * blockDim.x + threadIdx.x;
    if (g < N_GRAPHS) out[g] = head_b[0];
}

__global__ __launch_bounds__(256) void k_pool_head(const float* __restrict__ h,
                                                   const int* __restrict__ batch,
                                                   const float* __restrict__ hw,
                                                   float* __restrict__ out) {
    int node = blockIdx.x * 8 + (threadIdx.x >> 5);
    if (node >= N_NODES) return;
    int lane = threadIdx.x & 31;
    float4 v = ((const float4*)(h + (size_t)node * CH))[lane];
    float4 w = ((const float4*)hw)[lane];
    float dot = v.x * w.x + v.y * w.y + v.z * w.z + v.w * w.w;
#pragma unroll
    for (int off = 16; off > 0; off >>= 1) dot += __shfl_xor(dot, off, 32);
    if (lane == 0) atomAddF(&out[batch[node]], dot);
}

// ---------------------------------------------------------------- launcher
extern "C" void kernel_launch(void* const* d_in, const int* in_sizes, int n_in,
                              void* d_out, int out_size, void* d_ws, size_t ws_size,
                              hipStream_t stream) {
    const float* x      = (const float*)d_in[0];
    const int*   ei     = (const int*)d_in[1];          // [2][E]: row0=src, row1=dst
    const int*   batch  = (const int*)d_in[2];
    const float* Ws     = (const float*)d_in[3];        // [3][128][128]
    const float* bs     = (const float*)d_in[4];        // [3][128]
    const float* head_w = (const float*)d_in[5];        // [128]
    const float* head_b = (const float*)d_in[6];        // [1]
    float*       out    = (float*)d_out;

    const int* src = ei;
    const int* dst = ei + N_EDGES;

    float* dinv = (float*)d_ws;                         // N floats
    float* t    = dinv + N_NODES;                       // N*128 floats (16B-aligned)
    float* h    = t + (size_t)N_NODES * CH;             // N*128 floats

    // normalization: deg over dst with self-loops, dinv = deg^-1/2
    k_init_deg <<<(N_NODES + 255) / 256, 256, 0, stream>>>(dinv);
    k_deg_edges<<<N_EDGES / 256,        256, 0, stream>>>(dst, dinv);
    k_make_dinv<<<(N_NODES + 255) / 256, 256, 0, stream>>>(dinv);

    const int elem4Blocks = (N_NODES * (CH / 4)) / 256; // 12500 exact
    for (int l = 0; l < N_LAYERS; ++l) {
        const float* hin = (l == 0) ? x : h;
        k_gemm_wmma<<<(N_NODES + 127) / 128, 256, 0, stream>>>(
            hin, Ws + (size_t)l * CH * CH, t, N_NODES);
        k_self_init<<<elem4Blocks, 256, 0, stream>>>(t, dinv, h);
        k_scatter  <<<N_EDGES / 8, 256, 0, stream>>>(src, dst, dinv, t, h);
        k_bias_relu<<<elem4Blocks, 256, 0, stream>>>(h, bs + (size_t)l * CH);
    }

    k_out_init <<<(N_GRAPHS + 255) / 256, 256, 0, stream>>>(out, head_b);
    k_pool_head<<<N_NODES / 8, 256, 0, stream>>>(h, batch, head_w, out);
}